// MoeLayer_67061619359881
// MI455X (gfx1250) — compile-verified
//
#include <hip/hip_runtime.h>
#include <math.h>

typedef float v2f __attribute__((ext_vector_type(2)));
typedef float v8f __attribute__((ext_vector_type(8)));

#define NE 8
#define NK 2
#define ND 1024
#define NB 4
#define NS 2048
#define NT 8192      // total tokens B*S
#define TM 32        // token-tile rows per block
#define SLAB 128     // k-slab staged in LDS for GEMM1
#define LXS 132      // lx row stride in floats (132 % 64 == 4 -> conflict-free A-frag reads)
#define LHS 1028     // lh row stride in floats (1028 % 64 == 4)

__device__ __forceinline__ float gelu_exact(float v) {
    return 0.5f * v * (1.0f + erff(v * 0.70710678118654752f));
}

// ---------------- gating: logits, softmax, top-2, loss partials ----------------
__global__ void gate_kernel(const float* __restrict__ x, const float* __restrict__ Wg,
                            float* __restrict__ p_all, float* __restrict__ zsq,
                            int* __restrict__ sel1, int* __restrict__ sel2,
                            float* __restrict__ c1, float* __restrict__ c2,
                            int* __restrict__ cnt, int* __restrict__ d1cnt) {
    __shared__ float wg[NE * ND];
    int tid = threadIdx.x;
    for (int i = tid; i < NE * ND / 4; i += 256)
        ((float4*)wg)[i] = ((const float4*)Wg)[i];
    __syncthreads();

    int wv = tid >> 5, lane = tid & 31;
    int t = blockIdx.x * 8 + wv;                 // one wave per token
    const float* xr = x + (size_t)t * ND;

    float acc[NE];
    #pragma unroll
    for (int e = 0; e < NE; ++e) acc[e] = 0.f;
    for (int d = lane; d < ND; d += 32) {
        float xv = xr[d];
        #pragma unroll
        for (int e = 0; e < NE; ++e) acc[e] += xv * wg[e * ND + d];
    }
    #pragma unroll
    for (int e = 0; e < NE; ++e) {
        float v = acc[e];
        v += __shfl_xor(v, 16, 32);
        v += __shfl_xor(v, 8, 32);
        v += __shfl_xor(v, 4, 32);
        v += __shfl_xor(v, 2, 32);
        v += __shfl_xor(v, 1, 32);
        acc[e] = v;
    }
    if (lane == 0) {
        float m = acc[0];
        #pragma unroll
        for (int e = 1; e < NE; ++e) m = fmaxf(m, acc[e]);
        float s = 0.f, p[NE];
        #pragma unroll
        for (int e = 0; e < NE; ++e) { p[e] = expf(acc[e] - m); s += p[e]; }
        float inv = 1.f / s;
        #pragma unroll
        for (int e = 0; e < NE; ++e) p[e] *= inv;
        float z = m + logf(s);                   // logsumexp

        int i1 = 0; float v1 = p[0];
        #pragma unroll
        for (int e = 1; e < NE; ++e) if (p[e] > v1) { v1 = p[e]; i1 = e; }
        int i2 = (i1 == 0) ? 1 : 0; float v2 = p[i2];
        #pragma unroll
        for (int e = 0; e < NE; ++e) if (e != i1 && p[e] > v2) { v2 = p[e]; i2 = e; }

        float wsum = v1 + v2;
        sel1[t] = i1; sel2[t] = i2;
        c1[t] = v1 / wsum; c2[t] = v2 / wsum;
        #pragma unroll
        for (int e = 0; e < NE; ++e) p_all[(size_t)t * NE + e] = p[e];
        zsq[t] = z * z;
        atomicAdd(&cnt[i1], 1);
        atomicAdd(&cnt[i2], 1);
        int b = t >> 11;                          // t / S
        atomicAdd(&d1cnt[b * NE + i1], 1);        // int atomics: deterministic
    }
}

// ---------------- deterministic reductions ----------------
__global__ void reduce_zsq(const float* __restrict__ zsq, float* __restrict__ ztot) {
    __shared__ float red[256];
    int tid = threadIdx.x;
    float s = 0.f;
    for (int i = tid; i < NT; i += 256) s += zsq[i];     // fixed order
    red[tid] = s; __syncthreads();
    for (int o = 128; o > 0; o >>= 1) { if (tid < o) red[tid] += red[tid + o]; __syncthreads(); }
    if (tid == 0) ztot[0] = red[0];
}

__global__ void reduce_proxy(const float* __restrict__ p_all, float* __restrict__ proxy) {
    __shared__ float red[256];
    int be = blockIdx.x; int b = be >> 3, e = be & 7;    // one block per (b,e)
    int tid = threadIdx.x;
    float s = 0.f;
    for (int sidx = tid; sidx < NS; sidx += 256)
        s += p_all[((size_t)(b * NS + sidx)) * NE + e];
    red[tid] = s; __syncthreads();
    for (int o = 128; o > 0; o >>= 1) { if (tid < o) red[tid] += red[tid + o]; __syncthreads(); }
    if (tid == 0) proxy[be] = red[0];
}

// ---------------- expert-list construction ----------------
__global__ void scan_kernel(const int* __restrict__ cnt, int* __restrict__ off) {
    if (threadIdx.x == 0 && blockIdx.x == 0) {
        int r = 0;
        for (int e = 0; e < NE; ++e) { off[e] = r; r += cnt[e]; }
    }
}

__global__ void build_kernel(const int* __restrict__ sel1, const int* __restrict__ sel2,
                             const float* __restrict__ c1, const float* __restrict__ c2,
                             const int* __restrict__ off, int* __restrict__ cursor,
                             int* __restrict__ rows, float* __restrict__ rwt) {
    int t = blockIdx.x * 256 + threadIdx.x;
    if (t >= NT) return;
    int e1 = sel1[t]; int p1 = atomicAdd(&cursor[e1], 1);
    rows[off[e1] + p1] = t; rwt[off[e1] + p1] = c1[t];
    int e2 = sel2[t]; int p2 = atomicAdd(&cursor[e2], 1);
    rows[off[e2] + p2] = t; rwt[off[e2] + p2] = c2[t];
}

// ---------------- fused expert FFN: GEMM1 -> GELU -> GEMM2 (f32 WMMA) ----------------
__global__ void __launch_bounds__(256)
expert_kernel(const float* __restrict__ x,
              const float* __restrict__ W1, const float* __restrict__ b1,
              const float* __restrict__ W2, const float* __restrict__ b2,
              const int* __restrict__ cnt, const int* __restrict__ off,
              const int* __restrict__ rows, const float* __restrict__ rwt,
              float* __restrict__ out) {
    __shared__ float lx[TM * LXS];   // x k-slab (32 x 128, padded)
    __shared__ float lh[TM * LHS];   // full hidden tile (32 x 1024, padded)
    __shared__ int   ltok[TM];
    __shared__ float lwt[TM];

    int e    = blockIdx.x >> 8;      // 256 tiles per expert
    int tile = blockIdx.x & 255;
    int Mc = cnt[e];
    int M0 = tile * TM;
    if (M0 >= Mc) return;            // uniform early-exit
    int Mv = min(TM, Mc - M0);

    int tid = threadIdx.x;
    int wv = tid >> 5, lane = tid & 31;

    if (tid < TM) {
        int m = M0 + tid;
        if (m < Mc) { ltok[tid] = rows[off[e] + m]; lwt[tid] = rwt[off[e] + m]; }
        else        { ltok[tid] = 0;                lwt[tid] = 0.f; }
    }
    __syncthreads();

    const float* W1e = W1 + (size_t)e * ND * ND;
    const float* W2e = W2 + (size_t)e * ND * ND;

    int arow = lane & 15;                 // A-frag row within 16-row tile
    int koff = (lane >> 4) ? 2 : 0;       // hi half-wave holds K+2,K+3
    int ncol = lane & 15;                 // B/C column within 16-col tile

    const v8f zero8 = {0.f,0.f,0.f,0.f,0.f,0.f,0.f,0.f};
    v8f acc[8][2];                        // [col-tile][row-tile] -> 32x1024/8waves
    #pragma unroll
    for (int ct = 0; ct < 8; ++ct) { acc[ct][0] = zero8; acc[ct][1] = zero8; }

    // ---- GEMM1: h_pre = x_tile @ W1e ----
    for (int ks = 0; ks < ND; ks += SLAB) {
        {   // stage x slab: each thread copies 16 floats of one row
            int r  = tid >> 3;
            int c0 = (tid & 7) * 16;
            float* dst = lx + r * LXS + c0;
            if (r < Mv) {
                const float* src = x + (size_t)ltok[r] * ND + ks + c0;
                #pragma unroll
                for (int i = 0; i < 4; ++i) ((float4*)dst)[i] = ((const float4*)src)[i];
            } else {
                #pragma unroll
                for (int i = 0; i < 4; ++i) ((float4*)dst)[i] = make_float4(0.f,0.f,0.f,0.f);
            }
        }
        __syncthreads();
        for (int k0 = 0; k0 < SLAB; k0 += 4) {
            v2f a0, a1;
            a0.x = lx[arow * LXS + k0 + koff];
            a0.y = lx[arow * LXS + k0 + koff + 1];
            a1.x = lx[(arow + 16) * LXS + k0 + koff];
            a1.y = lx[(arow + 16) * LXS + k0 + koff + 1];
            int krow = ks + k0 + koff;
            #pragma unroll
            for (int ct = 0; ct < 8; ++ct) {
                int n = ct * 128 + wv * 16 + ncol;
                v2f bf;
                bf.x = W1e[(size_t)krow * ND + n];
                bf.y = W1e[(size_t)(krow + 1) * ND + n];
                acc[ct][0] = __builtin_amdgcn_wmma_f32_16x16x4_f32(
                    false, a0, false, bf, (short)0, acc[ct][0], false, false);
                acc[ct][1] = __builtin_amdgcn_wmma_f32_16x16x4_f32(
                    false, a1, false, bf, (short)0, acc[ct][1], false, false);
            }
        }
        __syncthreads();
    }

    // ---- bias + exact GELU -> lh ----
    #pragma unroll
    for (int ct = 0; ct < 8; ++ct) {
        int n = ct * 128 + wv * 16 + ncol;
        float bias = b1[e * ND + n];
        #pragma unroll
        for (int rt = 0; rt < 2; ++rt) {
            #pragma unroll
            for (int j = 0; j < 8; ++j) {
                int r = rt * 16 + ((lane < 16) ? j : j + 8);   // C-matrix VGPR layout
                lh[r * LHS + n] = gelu_exact(acc[ct][rt][j] + bias);
            }
        }
    }
    __syncthreads();

    // ---- GEMM2: y = h @ W2e ----
    #pragma unroll
    for (int ct = 0; ct < 8; ++ct) { acc[ct][0] = zero8; acc[ct][1] = zero8; }

    for (int k0 = 0; k0 < ND; k0 += 4) {
        v2f a0, a1;
        a0.x = lh[arow * LHS + k0 + koff];
        a0.y = lh[arow * LHS + k0 + koff + 1];
        a1.x = lh[(arow + 16) * LHS + k0 + koff];
        a1.y = lh[(arow + 16) * LHS + k0 + koff + 1];
        int krow = k0 + koff;
        #pragma unroll
        for (int ct = 0; ct < 8; ++ct) {
            int n = ct * 128 + wv * 16 + ncol;
            v2f bf;
            bf.x = W2e[(size_t)krow * ND + n];
            bf.y = W2e[(size_t)(krow + 1) * ND + n];
            acc[ct][0] = __builtin_amdgcn_wmma_f32_16x16x4_f32(
                false, a0, false, bf, (short)0, acc[ct][0], false, false);
            acc[ct][1] = __builtin_amdgcn_wmma_f32_16x16x4_f32(
                false, a1, false, bf, (short)0, acc[ct][1], false, false);
        }
    }

    // ---- weighted scatter (each token gets exactly 2 commutative adds) ----
    #pragma unroll
    for (int ct = 0; ct < 8; ++ct) {
        int n = ct * 128 + wv * 16 + ncol;
        float bias = b2[e * ND + n];
        #pragma unroll
        for (int rt = 0; rt < 2; ++rt) {
            #pragma unroll
            for (int j = 0; j < 8; ++j) {
                int r = rt * 16 + ((lane < 16) ? j : j + 8);
                if (r < Mv) {
                    float v = (acc[ct][rt][j] + bias) * lwt[r];
                    atomicAdd(&out[(size_t)ltok[r] * ND + n], v);
                }
            }
        }
    }
}

// ---------------- final scalar losses ----------------
__global__ void loss_final(const float* __restrict__ proxy, const int* __restrict__ d1cnt,
                           const float* __restrict__ ztot, float* __restrict__ out) {
    if (threadIdx.x == 0 && blockIdx.x == 0) {
        float bsum = 0.f;
        for (int i = 0; i < NB * NE; ++i)
            bsum += (proxy[i] / (float)NS) * ((float)d1cnt[i] / (float)NS);
        float balance = (bsum / (float)(NB * NE)) * (float)(NE * NE);
        float zloss = ztot[0] / (float)NT;
        out[(size_t)NT * ND + 0] = balance * 0.01f + zloss * 0.001f;
        out[(size_t)NT * ND + 1] = balance;
    }
}

extern "C" void kernel_launch(void* const* d_in, const int* in_sizes, int n_in,
                              void* d_out, int out_size, void* d_ws, size_t ws_size,
                              hipStream_t stream) {
    (void)in_sizes; (void)n_in; (void)ws_size;
    const float* x  = (const float*)d_in[0];
    const float* Wg = (const float*)d_in[1];
    const float* W1 = (const float*)d_in[2];
    const float* b1 = (const float*)d_in[3];
    const float* W2 = (const float*)d_in[4];
    const float* b2 = (const float*)d_in[5];
    float* out = (float*)d_out;

    char* w = (char*)d_ws;
    int*   cnt    = (int*)(w + 0);     // 8
    int*   cursor = (int*)(w + 64);    // 8
    int*   d1cnt  = (int*)(w + 128);   // 32
    int*   off    = (int*)(w + 256);   // 8
    float* ztot   = (float*)(w + 320); // 1
    float* proxy  = (float*)(w + 384); // 32
    size_t o = 512;
    int*   sel1  = (int*)(w + o);   o += (size_t)NT * 4;
    int*   sel2  = (int*)(w + o);   o += (size_t)NT * 4;
    float* c1    = (float*)(w + o); o += (size_t)NT * 4;
    float* c2    = (float*)(w + o); o += (size_t)NT * 4;
    float* p_all = (float*)(w + o); o += (size_t)NT * NE * 4;
    float* zsq   = (float*)(w + o); o += (size_t)NT * 4;
    int*   rowsL = (int*)(w + o);   o += (size_t)NT * NK * 4;
    float* rwtL  = (float*)(w + o); o += (size_t)NT * NK * 4;

    hipMemsetAsync(d_out, 0, (size_t)out_size * sizeof(float), stream);
    hipMemsetAsync(d_ws, 0, 512, stream);

    gate_kernel<<<NT / 8, 256, 0, stream>>>(x, Wg, p_all, zsq, sel1, sel2, c1, c2, cnt, d1cnt);
    reduce_zsq<<<1, 256, 0, stream>>>(zsq, ztot);
    reduce_proxy<<<NB * NE, 256, 0, stream>>>(p_all, proxy);
    scan_kernel<<<1, 1, 0, stream>>>(cnt, off);
    build_kernel<<<NT / 256, 256, 0, stream>>>(sel1, sel2, c1, c2, off, cursor, rowsL, rwtL);
    expert_kernel<<<NE * 256, 256, 0, stream>>>(x, W1, b1, W2, b2, cnt, off, rowsL, rwtL, out);
    loss_final<<<1, 1, 0, stream>>>(proxy, d1cnt, ztot, out);
}